// Convolutionv2106Custom_21586505630268
// MI455X (gfx1250) — compile-verified
//
#include <hip/hip_runtime.h>
#include <cmath>

// Problem constants (from reference)
#define N_NODES_   12500
#define N_EDGES_   200000
#define MUL0_      64
#define MUL1_      32
#define NODE_DIM_  160      // MUL0 + 3*MUL1
#define W_NUMEL_   192      // 64+64+32+32
#define OUT_DIM_   384      // 64 + 32 + 192 + 96
#define FC_IN_     16
#define FC_HID_    64
#define TILE_      16       // edges per wave
#define WAVES_     8        // waves per block (wave32)
#define BLOCK_     256
#define N_TILES_   (N_EDGES_ / TILE_)   // 12500 exactly

// Manual LDS layout (floats) so fragment offsets fit the 16-bit DS imm field:
//   [0, 12288)            w2 fragments (48 KB)   -> imm offsets 0..48895
//   [12288, 13312)        w1 fragments (4 KB)    -> imm offsets 49152..53247
//   [13312, 13312+24576)  per-wave wbuf (96 KB)
#define W2_OFF_   0
#define W1_OFF_   (FC_HID_ * W_NUMEL_)            // 12288
#define WBUF_OFF_ (W1_OFF_ + FC_IN_ * FC_HID_)    // 13312
#define SMEM_FLOATS_ (WBUF_OFF_ + WAVES_ * TILE_ * W_NUMEL_)  // 37888

typedef float v2f __attribute__((ext_vector_type(2)));
typedef float v8f __attribute__((ext_vector_type(8)));

// D = A(16x4 f32) * B(4x16 f32) + C(16x16 f32), wave32 WMMA.
__device__ __forceinline__ v8f wmma4(v2f a, v2f b, v8f c) {
  return __builtin_amdgcn_wmma_f32_16x16x4_f32(
      /*neg_a=*/false, a, /*neg_b=*/false, b,
      /*c_mod=*/(short)0, c, /*reuse_a=*/false, /*reuse_b=*/false);
}

// Weights live in LDS pre-swizzled into WMMA B-fragment order:
//   frag f = kf*NFRAGN + nf, value index = lane*2 + j   (j in {0,1})
//   maps to linear (k, n): k = 4*kf + 2*(lane>>4) + j, n = 16*nf + (lane&15)
// so each lane reads its v2f fragment with ONE ds_load_b64 at
//   base(lane*8 bytes) + f*256 (immediate offset).
__device__ __forceinline__ int frag_swizzle_index(int k, int n, int nfragn) {
  const int kf = k >> 2;
  const int rem = k & 3;            // 2*hi + j
  const int hi = rem >> 1;
  const int j  = rem & 1;
  const int nf = n >> 4;
  const int m  = n & 15;
  const int ln = (hi << 4) + m;
  return (((kf * nfragn + nf) << 5) + ln) * 2 + j;
}

__global__ __launch_bounds__(BLOCK_) void tp_conv_kernel(
    const float* __restrict__ node_input,    // (N, 160)
    const int*   __restrict__ edge_src,      // (E,)
    const int*   __restrict__ edge_dst,      // (E,)
    const float* __restrict__ edge_attr,     // (E, 4)
    const float* __restrict__ edge_scalars,  // (E, 16)
    const float* __restrict__ fc_w1,         // (16, 64)
    const float* __restrict__ fc_w2,         // (64, 192)
    float*       __restrict__ out,           // (N, 384), pre-zeroed
    float silu_norm)
{
  __shared__ __align__(16) float smem[SMEM_FLOATS_];   // 148 KB total

  const int tid = threadIdx.x;
  // Stage FC weights into LDS in fragment-swizzled order (coalesced global reads).
  for (int i = tid; i < FC_IN_ * FC_HID_; i += BLOCK_) {
    const int k = i >> 6, n = i & 63;                    // w1 is 16x64
    smem[W1_OFF_ + frag_swizzle_index(k, n, 4)] = fc_w1[i];
  }
  for (int i = tid; i < FC_HID_ * W_NUMEL_; i += BLOCK_) {
    const int k = i / W_NUMEL_, n = i - k * W_NUMEL_;    // w2 is 64x192
    smem[W2_OFF_ + frag_swizzle_index(k, n, 12)] = fc_w2[i];
  }
  __syncthreads();   // only block-wide barrier; all remaining LDS use is wave-private

  const int wave = tid >> 5;
  const int lane = tid & 31;
  const int tile = blockIdx.x * WAVES_ + wave;
  if (tile >= N_TILES_) return;            // wave-uniform; no barriers below

  float* wv = smem + WBUF_OFF_ + wave * (TILE_ * W_NUMEL_);
  const float* fragbase = smem + lane * 2; // single address reg; frag -> imm offset
  const int e0  = tile * TILE_;
  const int m16 = lane & 15;   // matrix row within half-wave
  const int hi  = lane >> 4;   // half-wave select
  const int klo = hi << 1;     // K offset {0,2} per f32 A/B fragment layout

  // ---------------- FC1: h = silu(scalars @ w1 / 4) * silu_norm ----------------
  v2f a1[4];
  {
    const float* srow = edge_scalars + (size_t)(e0 + m16) * FC_IN_ + klo;
    #pragma unroll
    for (int kf = 0; kf < 4; ++kf)
      a1[kf] = *(const v2f*)(srow + 4 * kf);   // K = 4*kf + klo, +1
  }
  #pragma unroll
  for (int nf = 0; nf < 4; ++nf) {             // 4 tiles of 16 output cols
    v8f acc = {};
    #pragma unroll
    for (int kf = 0; kf < 4; ++kf) {
      const v2f b = *(const v2f*)(fragbase + W1_OFF_ + ((kf * 4 + nf) << 6));
      acc = wmma4(a1[kf], b, acc);             // ds_load_b64 + 16-bit imm offset
    }
    #pragma unroll
    for (int r = 0; r < 8; ++r) {              // C row M = r + 8*hi
      float v = acc[r] * 0.25f;                // / sqrt(FC_IN)
      float s = (v / (1.0f + __expf(-v))) * silu_norm;
      wv[(r + (hi << 3)) * FC_HID_ + nf * 16 + m16] = s;  // h: 16x64 row-major
    }
  }

  // Re-fragment h (16x64) from LDS into 16 A fragments (same wave: LDS in-order).
  v2f a2[16];
  #pragma unroll
  for (int kf = 0; kf < 16; ++kf)
    a2[kf] = *(const v2f*)&wv[m16 * FC_HID_ + 4 * kf + klo];

  // ---------------- FC2: w = h @ w2 / 8 ----------------
  #pragma unroll
  for (int nf = 0; nf < 12; ++nf) {            // 12 tiles of 16 output cols
    v8f acc = {};
    #pragma unroll
    for (int kf = 0; kf < 16; ++kf) {
      const v2f b = *(const v2f*)(fragbase + W2_OFF_ + ((kf * 12 + nf) << 6));
      acc = wmma4(a2[kf], b, acc);             // ds_load_b64 + 16-bit imm offset
    }
    #pragma unroll
    for (int r = 0; r < 8; ++r)
      wv[(r + (hi << 3)) * W_NUMEL_ + nf * 16 + m16] = acc[r] * 0.125f; // /sqrt(64)
  }

  // ---------------- tensor product + scatter-add ----------------
  // out layout per edge: [out_a(64) | out_d(32) | out_b(64x3) | out_c(32x3)]
  // w layout per edge:   [w_a(64) | w_b(64) | w_c(32) | w_d(32)]
  for (int e = 0; e < TILE_; ++e) {
    const int ge  = e0 + e;
    const int src = edge_src[ge];
    const int dst = edge_dst[ge];
    if (e + 1 < TILE_) {  // prefetch next edge's gathered row (lanes span 640 B)
      const int srcn = edge_src[ge + 1];
      __builtin_prefetch(node_input + (size_t)srcn * NODE_DIM_ + lane * 5, 0, 0);
    }
    const float4 ea = *(const float4*)(edge_attr + (size_t)ge * 4);
    const float y0 = ea.x, y1x = ea.y, y1y = ea.z, y1z = ea.w;
    const float* xrow = node_input + (size_t)src * NODE_DIM_;
    float*       orow = out        + (size_t)dst * OUT_DIM_;
    const float* wrow = wv + e * W_NUMEL_;
    #pragma unroll
    for (int j = 0; j < 12; ++j) {             // 12*32 = 384 channels
      const int ch = j * 32 + lane;            // each j stays in one segment
      float val;
      if (ch < 64) {                           // out_a = w_a * x0 * y0
        val = wrow[ch] * xrow[ch] * y0;
      } else if (ch < 96) {                    // out_d = w_d * (x1 . y1) / sqrt(3)
        const int u = ch - 64;
        const float d = xrow[64 + 3*u] * y1x + xrow[65 + 3*u] * y1y
                      + xrow[66 + 3*u] * y1z;
        val = wrow[160 + u] * d * 0.57735026919f;
      } else if (ch < 288) {                   // out_b = (w_b * x0) outer y1
        const int idx = ch - 96;
        const int u = idx / 3;
        const int m = idx - 3 * u;
        const float ym = (m == 0) ? y1x : ((m == 1) ? y1y : y1z);
        val = wrow[64 + u] * xrow[u] * ym;
      } else {                                 // out_c = w_c * x1 * y0
        const int idx = ch - 288;
        const int u = idx / 3;
        val = wrow[128 + u] * xrow[64 + idx] * y0;
      }
      atomicAdd(orow + ch, val * 0.25f);       // / sqrt(NUM_NEIGHBORS)
    }
  }
}

// Replicates the reference's np.trapz-based SiLU 2nd-moment normalizer (host, f64).
static float compute_silu_norm() {
  const int n = 200001;
  const double z0 = -12.0, dz = 24.0 / (n - 1);
  const double inv_sqrt_2pi = 0.3989422804014327;
  double acc = 0.0, prev = 0.0;
  for (int i = 0; i < n; ++i) {
    double z = z0 + dz * i;
    double pdf = exp(-0.5 * z * z) * inv_sqrt_2pi;
    double s = z / (1.0 + exp(-z));
    double f = s * s * pdf;
    if (i) acc += 0.5 * (prev + f) * dz;
    prev = f;
  }
  return (float)(1.0 / sqrt(acc));
}

extern "C" void kernel_launch(void* const* d_in, const int* in_sizes, int n_in,
                              void* d_out, int out_size, void* d_ws, size_t ws_size,
                              hipStream_t stream) {
  const float* node_input   = (const float*)d_in[0];
  const int*   edge_src     = (const int*)  d_in[1];
  const int*   edge_dst     = (const int*)  d_in[2];
  const float* edge_attr    = (const float*)d_in[3];
  const float* edge_scalars = (const float*)d_in[4];
  const float* fc_w1        = (const float*)d_in[5];
  const float* fc_w2        = (const float*)d_in[6];
  float* out = (float*)d_out;

  // Output is accumulated with atomics; zero it first (capturable stream op).
  hipMemsetAsync(out, 0, (size_t)out_size * sizeof(float), stream);

  const float silu_norm = compute_silu_norm();
  const int grid = (N_TILES_ + WAVES_ - 1) / WAVES_;   // 1563 blocks
  tp_conv_kernel<<<grid, BLOCK_, 0, stream>>>(
      node_input, edge_src, edge_dst, edge_attr, edge_scalars,
      fc_w1, fc_w2, out, silu_norm);
}